// PQCAgent_64132451664143
// MI455X (gfx1250) — compile-verified
//
#include <hip/hip_runtime.h>
#include <math.h>

typedef float v2f __attribute__((ext_vector_type(2)));
typedef float v8f __attribute__((ext_vector_type(8)));

#define N_LAYERS 5
#define BETA 1.0f

// ---------------- complex helpers (precompute kernel) ----------------
struct cplx { float re, im; };
__device__ inline cplx cmul(cplx a, cplx b) {
    return { a.re * b.re - a.im * b.im, a.re * b.im + a.im * b.re };
}
__device__ inline cplx cadd(cplx a, cplx b) { return { a.re + b.re, a.im + b.im }; }

struct mat2 { cplx m[2][2]; };
__device__ inline mat2 mmul(const mat2& A, const mat2& B) {
    mat2 C;
#pragma unroll
    for (int i = 0; i < 2; ++i)
#pragma unroll
        for (int j = 0; j < 2; ++j)
            C.m[i][j] = cadd(cmul(A.m[i][0], B.m[0][j]), cmul(A.m[i][1], B.m[1][j]));
    return C;
}

// Build the 6 batch-independent 16x16 complex layer matrices:
//   l < 5 : diag(RING_PHASE) * (Rot_q0 x Rot_q1 x Rot_q2 x Rot_q3)
//   l = 5 : Kronecker product only (no ring)
// Qubit q corresponds to bit (3-q) of the flat state index (C-order tensor).
__global__ void pqc_precompute(const float* __restrict__ theta,
                               float* __restrict__ Lr, float* __restrict__ Li) {
    const int l  = blockIdx.x;     // 0..5
    const int t  = threadIdx.x;    // 0..255
    const int sp = t >> 4;         // output row s'
    const int s  = t & 15;         // input  col s

    mat2 M[4];
#pragma unroll
    for (int q = 0; q < 4; ++q) {
        const float p0 = theta[(l * 4 + q) * 3 + 0];
        const float p1 = theta[(l * 4 + q) * 3 + 1];
        const float p2 = theta[(l * 4 + q) * 3 + 2];
        // half-angles are in [0, pi/2]: hardware sin/cos path, no libm slow path
        float c0, s0, c1, s1, c2, s2;
        __sincosf(0.5f * p0, &s0, &c0);
        __sincosf(0.5f * p1, &s1, &c1);
        __sincosf(0.5f * p2, &s2, &c2);
        mat2 RX = { { { { c0, 0.f }, { 0.f, -s0 } }, { { 0.f, -s0 }, { c0, 0.f } } } };
        mat2 RY = { { { { c1, 0.f }, { -s1, 0.f } }, { { s1, 0.f }, { c1, 0.f } } } };
        mat2 RZ = { { { { c2, -s2 }, { 0.f, 0.f } }, { { 0.f, 0.f }, { c2, s2 } } } };
        M[q] = mmul(RZ, mmul(RY, RX));
    }

    float ring = 1.0f;
    if (l < N_LAYERS) {
        const int b0 = (sp >> 3) & 1, b1 = (sp >> 2) & 1, b2 = (sp >> 1) & 1, b3 = sp & 1;
        ring = ((b0 & b1) ^ (b1 & b2) ^ (b2 & b3) ^ (b3 & b0)) ? -1.0f : 1.0f;
    }
    cplx e = { ring, 0.f };
#pragma unroll
    for (int q = 0; q < 4; ++q) {
        const int bp = (sp >> (3 - q)) & 1;
        const int bs = (s  >> (3 - q)) & 1;
        e = cmul(e, M[q].m[bp][bs]);
    }
    Lr[l * 256 + sp * 16 + s] = e.re;
    Li[l * 256 + sp * 16 + s] = e.im;
}

// ---------------- main circuit kernel ----------------
__device__ inline v8f wmma4(v2f a, v2f b, v8f c) {
    // f32 WMMA: A/B NEG must be 0 (only CNeg legal per ISA 7.12)
    return __builtin_amdgcn_wmma_f32_16x16x4_f32(false, a, false, b, (short)0, c,
                                                 false, false);
}

// Apply 16x16 complex matrix L (row-major Re/Im tables) to the wave's state.
// State: R[8]/I[8] in WMMA C/D layout (vgpr r = rows r / r+8, col = lane%16).
__device__ inline void apply_matrix(const float* __restrict__ Lr,
                                    const float* __restrict__ Li,
                                    float R[8], float I[8],
                                    int row, bool hi) {
    v8f Rn = { 0, 0, 0, 0, 0, 0, 0, 0 };
    v8f In = { 0, 0, 0, 0, 0, 0, 0, 0 };
    const int aoff = row * 16 + (hi ? 2 : 0);
#pragma unroll
    for (int k = 0; k < 4; ++k) {
        const int base = 4 * (k & 1);
        // A chunk: columns 4k..4k+3 of L (ISA A-layout: K=0/1 lo-half, K=2/3 hi-half)
        v2f ar = *(const v2f*)(Lr + aoff + 4 * k);
        v2f ai = *(const v2f*)(Li + aoff + 4 * k);
        // B chunk: state rows 4k..4k+3 (ISA B-layout: vgpr j = K=j lo / K=j+2 hi)
        float shx_r = __shfl_xor((k < 2) ? R[base + 2] : R[base + 0], 16, 32);
        float shy_r = __shfl_xor((k < 2) ? R[base + 3] : R[base + 1], 16, 32);
        float shx_i = __shfl_xor((k < 2) ? I[base + 2] : I[base + 0], 16, 32);
        float shy_i = __shfl_xor((k < 2) ? I[base + 3] : I[base + 1], 16, 32);
        v2f br, bi;
        if (k < 2) {
            br.x = hi ? shx_r : R[base + 0];  br.y = hi ? shy_r : R[base + 1];
            bi.x = hi ? shx_i : I[base + 0];  bi.y = hi ? shy_i : I[base + 1];
        } else {
            br.x = hi ? R[base + 2] : shx_r;  br.y = hi ? R[base + 3] : shy_r;
            bi.x = hi ? I[base + 2] : shx_i;  bi.y = hi ? I[base + 3] : shy_i;
        }
        v2f nbi = -bi;
        Rn = wmma4(ar, br, Rn);   // + Lr*Sr
        Rn = wmma4(ai, nbi, Rn);  // - Li*Si
        In = wmma4(ar, bi, In);   // + Lr*Si
        In = wmma4(ai, br, In);   // + Li*Sr
    }
#pragma unroll
    for (int r = 0; r < 8; ++r) { R[r] = Rn[r]; I[r] = In[r]; }
}

__global__ __launch_bounds__(256)
void pqc_main(const float* __restrict__ x,
              const float* __restrict__ lmbd,
              const float* __restrict__ w,
              const float* __restrict__ Lr,
              const float* __restrict__ Li,
              float* __restrict__ out, int batch) {
    const int tid  = blockIdx.x * blockDim.x + threadIdx.x;
    const int lane = threadIdx.x & 31;
    const int col  = lane & 15;
    const bool hi  = (lane & 16) != 0;
    const int wave = tid >> 5;
    const int b    = wave * 16 + col;
    const int bb   = (b < batch) ? b : (batch - 1);   // clamp loads, keep EXEC full
    const int row  = col;                             // A-matrix row for this lane

    const float4 xb = *(const float4*)(x + bb * 4);   // x[b][0..3]
    const float xq[4] = { xb.x, xb.y, xb.z, xb.w };

    float R[8], I[8];
#pragma unroll
    for (int r = 0; r < 8; ++r) { R[r] = 0.f; I[r] = 0.f; }
    if (!hi) R[0] = 1.0f;   // |0000> amplitude = 1 in every batch column

#pragma unroll 1
    for (int l = 0; l < N_LAYERS; ++l) {
        // batch-independent rotations + ring phase via WMMA
        apply_matrix(Lr + l * 256, Li + l * 256, R, I, row, hi);
        // batch-dependent RX gates (angle = lmbd[4l+q] * x[b,q]); qubit q = bit (3-q)
#pragma unroll
        for (int q = 0; q < 4; ++q) {
            const float th = 0.5f * lmbd[4 * l + q] * xq[q];
            // angles are O(1): hardware sin/cos path, no libm slow-path branches
            float sn, cs;
            __sincosf(th, &sn, &cs);
            const int p = 3 - q;  // bit position mixed by this gate
            if (p == 3) {
                // rows s <-> s+8 : cross lane-half
#pragma unroll
                for (int r = 0; r < 8; ++r) {
                    const float shR = __shfl_xor(R[r], 16, 32);
                    const float shI = __shfl_xor(I[r], 16, 32);
                    const float nR = cs * R[r] + sn * shI;
                    const float nI = cs * I[r] - sn * shR;
                    R[r] = nR; I[r] = nI;
                }
            } else {
#pragma unroll
                for (int r0 = 0; r0 < 8; ++r0) {
                    if (r0 & (1 << p)) continue;
                    const int r1 = r0 | (1 << p);
                    const float Ra = R[r0], Ia = I[r0];
                    const float Rb = R[r1], Ib = I[r1];
                    R[r0] = cs * Ra + sn * Ib;  I[r0] = cs * Ia - sn * Rb;
                    R[r1] = cs * Rb + sn * Ia;  I[r1] = cs * Ib - sn * Ra;
                }
            }
        }
    }
    // final rotation layer (no ring, no RX)
    apply_matrix(Lr + N_LAYERS * 256, Li + N_LAYERS * 256, R, I, row, hi);

    // expectation of parity: sum_s (-1)^popc(s) * |amp_s|^2
    float partial = 0.f;
#pragma unroll
    for (int r = 0; r < 8; ++r) {
        const float m2 = R[r] * R[r] + I[r] * I[r];
        partial += (__popc(r) & 1) ? -m2 : m2;
    }
    if (hi) partial = -partial;                 // rows r+8 flip parity
    const float e = partial + __shfl_xor(partial, 16, 32);

    if (!hi && b < batch) {
        const float w0 = w[0], w1 = w[1];
        const float l0 = BETA * e * w0, l1 = BETA * e * w1;
        const float m  = fmaxf(l0, l1);
        const float e0 = __expf(l0 - m), e1 = __expf(l1 - m);
        const float inv = 1.0f / (e0 + e1);
        float2 pr = { e0 * inv, e1 * inv };
        *(float2*)(out + 2 * b) = pr;
    }
}

// ---------------- launcher ----------------
extern "C" void kernel_launch(void* const* d_in, const int* in_sizes, int n_in,
                              void* d_out, int out_size, void* d_ws, size_t ws_size,
                              hipStream_t stream) {
    const float* x     = (const float*)d_in[0];
    const float* theta = (const float*)d_in[1];
    const float* lmbd  = (const float*)d_in[2];
    const float* w     = (const float*)d_in[3];
    float* out = (float*)d_out;

    float* Lr = (float*)d_ws;          // 6 * 256 floats
    float* Li = Lr + 6 * 256;          // 6 * 256 floats  (12 KB total)

    const int batch = in_sizes[0] / 4;

    pqc_precompute<<<6, 256, 0, stream>>>(theta, Lr, Li);

    const int waves   = (batch + 15) / 16;
    const int threads = waves * 32;
    const int block   = 256;
    const int grid    = (threads + block - 1) / block;
    pqc_main<<<grid, block, 0, stream>>>(x, lmbd, w, Lr, Li, out, batch);
}